// GCNLayer_52329881534832
// MI455X (gfx1250) — compile-verified
//
#include <hip/hip_runtime.h>

typedef __attribute__((ext_vector_type(2))) float v2f;
typedef __attribute__((ext_vector_type(4))) float v4f;
typedef __attribute__((ext_vector_type(8))) float v8f;
typedef int v4i __attribute__((vector_size(16)));
typedef __attribute__((address_space(1))) v4i g_v4i;   // global int4
typedef __attribute__((address_space(3))) v4i l_v4i;   // LDS int4

#define N_NODES   100000
#define N_EDGES   1600000
#define IN_FEATS  256
#define OUT_FEATS 64
#define ROW_TILES (N_NODES / 16)   // 6250 exact
#define WAVES_PER_BLOCK 8
#define W_ELEMS   (IN_FEATS * OUT_FEATS)    // 16384 floats = 64KB

// ---------------------------------------------------------------------------
// Zero the accumulator (d_out) — must run every call (graph replay safe).
// ---------------------------------------------------------------------------
__global__ void gcn_zero(float* __restrict__ out, int n) {
    int i = blockIdx.x * blockDim.x + threadIdx.x;
    if (i < n) out[i] = 0.0f;
}

// ---------------------------------------------------------------------------
// h = x @ W  via V_WMMA_F32_16X16X4_F32 (fp32 WMMA, full precision).
// W (64KB) staged once per workgroup into LDS via async-to-LDS copies
// (ASYNCcnt), then the K-loop reads B fragments from LDS (ds path) while
// the VMEM path streams x. One wave computes a 16x64 strip of h.
// ---------------------------------------------------------------------------
__global__ __launch_bounds__(256) void gcn_gemm_wmma(
        const float* __restrict__ x,      // [N_NODES, IN_FEATS]
        const float* __restrict__ w,      // [IN_FEATS, OUT_FEATS]
        float* __restrict__ h)            // [N_NODES, OUT_FEATS]
{
    __shared__ float Wlds[W_ELEMS];       // 64 KB (WGP has 320 KB)

    // ---- stage W -> LDS (whole block cooperates) -------------------------
#if __has_builtin(__builtin_amdgcn_global_load_async_to_lds_b128)
    {
        for (int i = threadIdx.x; i < W_ELEMS / 4; i += 256) {
            __builtin_amdgcn_global_load_async_to_lds_b128(
                (g_v4i*)(w + 4 * i),
                (l_v4i*)(Wlds + 4 * i),
                0, 0);
        }
    }
#if __has_builtin(__builtin_amdgcn_s_wait_asynccnt)
    __builtin_amdgcn_s_wait_asynccnt(0);
#else
    asm volatile("s_wait_asynccnt 0" ::: "memory");
#endif
#else
    {
        const v4f* wg = (const v4f*)w;
        v4f* wl = (v4f*)Wlds;
        for (int i = threadIdx.x; i < W_ELEMS / 4; i += 256)
            wl[i] = wg[i];
    }
#endif
    __syncthreads();                      // all 256 threads reach this

    const int lane = threadIdx.x & 31;
    const int wave = threadIdx.x >> 5;
    const int tile = blockIdx.x * WAVES_PER_BLOCK + wave;
    if (tile >= ROW_TILES) return;        // wave-uniform: EXEC all-1s inside

    // A 16x4 fp32 layout: lanes 0-15 -> M=lane, K={0,1}; lanes 16-31 -> K={2,3}
    const int m     = lane & 15;
    const int khalf = (lane >> 4) << 1;   // 0 or 2
    const int col   = lane & 15;          // B/C column = lane & 15

    const float* xrow = x + (size_t)(tile * 16 + m) * IN_FEATS + khalf;

    v8f c0 = {}, c1 = {}, c2 = {}, c3 = {};

    for (int k = 0; k < IN_FEATS; k += 4) {
        // A fragment: contiguous float2 per lane (global, HBM stream)
        v2f a = *(const v2f*)(xrow + k);

        // B 4x16 layout from LDS: b.x = W[k+khalf][col], b.y = W[k+khalf+1][col]
        const float* wk0 = Wlds + (k + khalf) * OUT_FEATS + col;
        const float* wk1 = wk0 + OUT_FEATS;
        v2f b0 = { wk0[0],  wk1[0]  };
        v2f b1 = { wk0[16], wk1[16] };
        v2f b2 = { wk0[32], wk1[32] };
        v2f b3 = { wk0[48], wk1[48] };

        c0 = __builtin_amdgcn_wmma_f32_16x16x4_f32(false, a, false, b0, (short)0, c0, false, false);
        c1 = __builtin_amdgcn_wmma_f32_16x16x4_f32(false, a, false, b1, (short)0, c1, false, false);
        c2 = __builtin_amdgcn_wmma_f32_16x16x4_f32(false, a, false, b2, (short)0, c2, false, false);
        c3 = __builtin_amdgcn_wmma_f32_16x16x4_f32(false, a, false, b3, (short)0, c3, false, false);
    }

    // C/D 16x16 layout: VGPR r -> M = r + 8*(lane>=16), N = lane&15
    const int mhi = (lane >> 4) << 3;     // 0 or 8
    float* hrow = h + (size_t)(tile * 16 + mhi) * OUT_FEATS + col;
    #pragma unroll
    for (int r = 0; r < 8; ++r) {
        hrow[(size_t)r * OUT_FEATS +  0] = c0[r];
        hrow[(size_t)r * OUT_FEATS + 16] = c1[r];
        hrow[(size_t)r * OUT_FEATS + 32] = c2[r];
        hrow[(size_t)r * OUT_FEATS + 48] = c3[r];
    }
}

// ---------------------------------------------------------------------------
// Edge aggregation: one wave per edge; each lane owns 2 of the 64 features.
// h (25.6MB) and agg (25.6MB) are L2-resident (192MB L2) -> L2 atomic bound.
// ---------------------------------------------------------------------------
__global__ __launch_bounds__(256) void gcn_edge_aggregate(
        const float* __restrict__ h,      // [N_NODES, OUT_FEATS]
        const int*   __restrict__ src,    // [N_EDGES]
        const int*   __restrict__ dst,    // [N_EDGES]
        const float* __restrict__ ew,     // [N_EDGES]
        float*       __restrict__ agg)    // [N_NODES, OUT_FEATS]
{
    const int lane = threadIdx.x & 31;
    const int e = (int)((blockIdx.x * (unsigned)blockDim.x + threadIdx.x) >> 5);
    if (e >= N_EDGES) return;

    const int   s   = src[e];
    const int   d   = dst[e];
    const float wgt = ew[e];

    // coalesced 256B gather of h[s] across the wave (float2 per lane)
    const v2f hv = *(const v2f*)(h + (size_t)s * OUT_FEATS + lane * 2);

    float* outp = agg + (size_t)d * OUT_FEATS + lane * 2;
    atomicAdd(outp + 0, hv.x * wgt);      // global_atomic_add_f32 (L2-resident)
    atomicAdd(outp + 1, hv.y * wgt);
}

// ---------------------------------------------------------------------------
// out = relu(out + bias), in place.
// ---------------------------------------------------------------------------
__global__ void gcn_bias_relu(float* __restrict__ out,
                              const float* __restrict__ bias, int n) {
    int i = blockIdx.x * blockDim.x + threadIdx.x;
    if (i < n) {
        float v = out[i] + bias[i & (OUT_FEATS - 1)];
        out[i] = v > 0.0f ? v : 0.0f;
    }
}

// ---------------------------------------------------------------------------
extern "C" void kernel_launch(void* const* d_in, const int* in_sizes, int n_in,
                              void* d_out, int out_size, void* d_ws, size_t ws_size,
                              hipStream_t stream) {
    const float* x    = (const float*)d_in[0];
    const int*   src  = (const int*)  d_in[1];
    const int*   dst  = (const int*)  d_in[2];
    const float* ew   = (const float*)d_in[3];
    const float* w    = (const float*)d_in[4];
    const float* bias = (const float*)d_in[5];

    float* out = (float*)d_out;              // agg accumulator, then final output
    float* h   = (float*)d_ws;               // 100000*64*4 = 25.6 MB scratch

    const int outN = N_NODES * OUT_FEATS;    // 6,400,000

    gcn_zero<<<(outN + 255) / 256, 256, 0, stream>>>(out, outN);

    gcn_gemm_wmma<<<(ROW_TILES + WAVES_PER_BLOCK - 1) / WAVES_PER_BLOCK, 256, 0, stream>>>(
        x, w, h);

    // one wave (32 threads) per edge -> N_EDGES*32 threads
    const long long eThreads = (long long)N_EDGES * 32;
    gcn_edge_aggregate<<<(int)((eThreads + 255) / 256), 256, 0, stream>>>(
        h, src, dst, ew, out);

    gcn_bias_relu<<<(outN + 255) / 256, 256, 0, stream>>>(out, bias, outN);
}